// WarpingLayer_64931315581436
// MI455X (gfx1250) — compile-verified
//
#include <hip/hip_runtime.h>
#include <math.h>

typedef __attribute__((ext_vector_type(2))) float v2f;
typedef __attribute__((ext_vector_type(8))) float v8f;

#define N_PTS 8192
#define N_B   4
#define BIGF  3.4e38f

// Branch-free sorted top-3 insert (s0 <= s1 <= s2).
__device__ __forceinline__ void top3_insert(float s, int idx,
                                            float& s0, float& s1, float& s2,
                                            int& i0, int& i1, int& i2) {
  bool l2 = s < s2, l1 = s < s1, l0 = s < s0;
  float ns2 = l1 ? s1 : (l2 ? s : s2);
  int   ni2 = l1 ? i1 : (l2 ? idx : i2);
  float ns1 = l0 ? s0 : (l1 ? s : s1);
  int   ni1 = l0 ? i0 : (l1 ? idx : i1);
  float ns0 = l0 ? s : s0;
  int   ni0 = l0 ? idx : i0;
  s0 = ns0; s1 = ns1; s2 = ns2; i0 = ni0; i1 = ni1; i2 = ni2;
}

// Pack warped candidates as A-operand rows (-2x,-2y,-2z,|c|^2) and flows.
__global__ __launch_bounds__(256) void prep_kernel(
    const float* __restrict__ xyz1, const float* __restrict__ flow1,
    float4* __restrict__ cand4, float4* __restrict__ flow4) {
  int tid = blockIdx.x * blockDim.x + threadIdx.x;   // [0, N_B*N_PTS)
  int b = tid >> 13;
  int n = tid & (N_PTS - 1);
  const float* xb = xyz1 + b * 3 * N_PTS;
  const float* fb = flow1 + b * 3 * N_PTS;
  float fx = fb[n], fy = fb[n + N_PTS], fz = fb[n + 2 * N_PTS];
  float x = xb[n] + fx, y = xb[n + N_PTS] + fy, z = xb[n + 2 * N_PTS] + fz;
  cand4[tid] = make_float4(-2.0f * x, -2.0f * y, -2.0f * z, x * x + y * y + z * z);
  flow4[tid] = make_float4(fx, fy, fz, 0.0f);
}

// One block per (batch, 16-query tile); 8 waves split the 8192 candidates.
__global__ __launch_bounds__(256) void knn_warp_kernel(
    const float* __restrict__ xyz2,
    const float4* __restrict__ cand4,
    const float4* __restrict__ flow4,
    float* __restrict__ out) {
  __shared__ float sS[8][16][3];
  __shared__ int   sI[8][16][3];

  const int lane = threadIdx.x & 31;
  const int wv   = threadIdx.x >> 5;
  const int hi   = lane >> 4;      // lane half selects K-pair {0,1} vs {2,3}
  const int lq   = lane & 15;      // query column / candidate row within tile

  const int b  = blockIdx.x >> 9;            // 512 query tiles per batch
  const int q0 = (blockIdx.x & 511) << 4;

  const float* x2 = xyz2 + b * 3 * N_PTS;
  const float qx = x2[q0 + lq];
  const float qy = x2[q0 + lq + N_PTS];
  const float qz = x2[q0 + lq + 2 * N_PTS];

  // B operand (4x16 f32): lanes 0-15 = (K0,K1)=(qx,qy); lanes 16-31 = (K2,K3)=(qz,1)
  v2f bm;
  bm.x = hi ? qz : qx;
  bm.y = hi ? 1.0f : qy;

  const float4* cb    = cand4 + b * N_PTS;
  const float*  cbase = (const float*)cb;

  float s0 = BIGF, s1 = BIGF, s2 = BIGF;
  int   i0 = 0, i1 = 0, i2 = 0;

  const int cstart = wv << 10;               // 1024 candidates per wave
  // Single base address; tiles advance by 16 candidates * 16 B = 256 B, so the
  // unrolled loads use one VGPR address pair + immediate offsets.
  const v2f* ap  = (const v2f*)(cbase + ((cstart + lq) << 2) + (hi << 1));
  const int ibase = cstart + (hi << 3);      // lane half covers rows 0-7 / 8-15

#pragma unroll 4
  for (int t = 0; t < 64; ++t) {
    // A operand (16x4 f32): lanes 0-15 = (-2x,-2y); lanes 16-31 = (-2z,|c|^2)
    v2f am = ap[t << 5];                     // one global_load_b64 per lane
    v8f acc = {};
    // score[m][n] = |c_m|^2 - 2 c_m . q_n   (rank-equivalent to sq distance)
    acc = __builtin_amdgcn_wmma_f32_16x16x4_f32(false, am, false, bm,
                                                (short)0, acc, false, false);
    const int base = ibase + (t << 4);
#pragma unroll
    for (int r = 0; r < 8; ++r)
      top3_insert(acc[r], base + r, s0, s1, s2, i0, i1, i2);
  }

  // Merge the two lane halves of each query column.
  float rs0 = __shfl_xor(s0, 16, 32);
  float rs1 = __shfl_xor(s1, 16, 32);
  float rs2 = __shfl_xor(s2, 16, 32);
  int   ri0 = __shfl_xor(i0, 16, 32);
  int   ri1 = __shfl_xor(i1, 16, 32);
  int   ri2 = __shfl_xor(i2, 16, 32);
  top3_insert(rs0, ri0, s0, s1, s2, i0, i1, i2);
  top3_insert(rs1, ri1, s0, s1, s2, i0, i1, i2);
  top3_insert(rs2, ri2, s0, s1, s2, i0, i1, i2);

  if (lane < 16) {
    sS[wv][lq][0] = s0; sS[wv][lq][1] = s1; sS[wv][lq][2] = s2;
    sI[wv][lq][0] = i0; sI[wv][lq][1] = i1; sI[wv][lq][2] = i2;
  }
  __syncthreads();

  if (threadIdx.x < 16) {                    // wave 0 lanes 0-15 finalize
    float t0 = BIGF, t1 = BIGF, t2 = BIGF;
    int   j0 = 0, j1 = 0, j2 = 0;
#pragma unroll
    for (int w = 0; w < 8; ++w)
#pragma unroll
      for (int k = 0; k < 3; ++k)
        top3_insert(sS[w][lq][k], sI[w][lq][k], t0, t1, t2, j0, j1, j2);

    const float4* fb = flow4 + b * N_PTS;
    int   jj[3] = {j0, j1, j2};
    float inv[3], gfx[3], gfy[3], gfz[3];
#pragma unroll
    for (int k = 0; k < 3; ++k) {
      float4 c = cb[jj[k]];                  // recover warped coords from pack
      float px = -0.5f * c.x, py = -0.5f * c.y, pz = -0.5f * c.z;
      float dx = px - qx, dy = py - qy, dz = pz - qz;
      float d = sqrtf(dx * dx + dy * dy + dz * dz);
      d = fmaxf(d, 1e-10f);
      inv[k] = 1.0f / d;
      float4 f = fb[jj[k]];
      gfx[k] = f.x; gfy[k] = f.y; gfz[k] = f.z;
    }
    float wr  = 1.0f / (inv[0] + inv[1] + inv[2]);
    float f2x = (inv[0] * gfx[0] + inv[1] * gfx[1] + inv[2] * gfx[2]) * wr;
    float f2y = (inv[0] * gfy[0] + inv[1] * gfy[1] + inv[2] * gfy[2]) * wr;
    float f2z = (inv[0] * gfz[0] + inv[1] * gfz[1] + inv[2] * gfz[2]) * wr;

    float* ob = out + b * 3 * N_PTS;
    int q = q0 + lq;
    ob[q]             = qx - f2x;
    ob[q + N_PTS]     = qy - f2y;
    ob[q + 2 * N_PTS] = qz - f2z;
  }
}

extern "C" void kernel_launch(void* const* d_in, const int* in_sizes, int n_in,
                              void* d_out, int out_size, void* d_ws, size_t ws_size,
                              hipStream_t stream) {
  (void)in_sizes; (void)n_in; (void)out_size; (void)ws_size;
  const float* xyz1  = (const float*)d_in[0];
  const float* xyz2  = (const float*)d_in[1];
  const float* flow1 = (const float*)d_in[2];
  float* out = (float*)d_out;

  float4* cand4 = (float4*)d_ws;                 // N_B*N_PTS float4 = 512 KB
  float4* flow4 = cand4 + N_B * N_PTS;           // + 512 KB

  prep_kernel<<<(N_B * N_PTS) / 256, 256, 0, stream>>>(xyz1, flow1, cand4, flow4);
  knn_warp_kernel<<<N_B * (N_PTS / 16), 256, 0, stream>>>(xyz2, cand4, flow4, out);
}